// NN_31095563223590
// MI455X (gfx1250) — compile-verified
//
#include <hip/hip_runtime.h>

typedef float v2f __attribute__((ext_vector_type(2)));
typedef float v8f __attribute__((ext_vector_type(8)));

#define B_DIM   16384
#define FEAT    1200
#define HID     256
#define CLS     10
#define NUM1    1000
#define NUM2    200
#define TOPK    200
#define SIM_LD  1024   // padded row stride for sim

__device__ __forceinline__ int imin(int a, int b) { return a < b ? a : b; }

// D = A*B + C  (f32, 16x16, K=4 per issue)
__device__ __forceinline__ v8f wmma_f32_k4(v2f a, v2f b, v8f c) {
    return __builtin_amdgcn_wmma_f32_16x16x4_f32(false, a, false, b, (short)0, c,
                                                 false, false);
}

// ---------------------------------------------------------------------------
// Kernel 1: sim[m][n] = (1/B) * sum_b x[b, fidx2[m]] * x[b, fidx1[n]]
// M=200 (7 tiles of 32), N=1000 (16 groups of 64), K=16384.
// Wave tile = 32(M) x 64(N): 2 A frags x 4 B frags -> 8 WMMA per K-step,
// 12 loads per 8 WMMA (1.5 loads/WMMA). Indices CLAMPED, not masked:
// padded rows/cols pollute only outputs the masked epilogue never stores.
// ---------------------------------------------------------------------------
__global__ __launch_bounds__(128) void sim_kernel(const float* __restrict__ x,
                                                  const int* __restrict__ fidx1,
                                                  const int* __restrict__ fidx2,
                                                  float* __restrict__ sim) {
    const int lane  = threadIdx.x & 31;
    const int wave  = threadIdx.x >> 5;
    const int l16   = lane & 15;
    const int half  = lane >> 4;
    const int mtile = blockIdx.y * 32;
    const int nbase = (blockIdx.x * 4 + wave) * 64;  // 4 N-tiles per wave

    const int acol0 = fidx2[imin(mtile + l16,      NUM2 - 1)];
    const int acol1 = fidx2[imin(mtile + 16 + l16, NUM2 - 1)];

    int bsrc[4];
#pragma unroll
    for (int t = 0; t < 4; ++t) {
        bsrc[t] = fidx1[imin(nbase + t * 16 + l16, NUM1 - 1)];
    }

    v8f acc[2][4] = {};
#pragma unroll 2
    for (int k0 = 0; k0 < B_DIM; k0 += 4) {
        const int ka = k0 + half * 2;
        const float* __restrict__ r0 = x + (size_t)ka * FEAT;
        const float* __restrict__ r1 = r0 + FEAT;
        v2f a0, a1;
        a0.x = r0[acol0]; a0.y = r1[acol0];
        a1.x = r0[acol1]; a1.y = r1[acol1];
        v2f b[4];
#pragma unroll
        for (int t = 0; t < 4; ++t) {
            b[t].x = r0[bsrc[t]];
            b[t].y = r1[bsrc[t]];
        }
#pragma unroll
        for (int t = 0; t < 4; ++t) {
            acc[0][t] = wmma_f32_k4(a0, b[t], acc[0][t]);
            acc[1][t] = wmma_f32_k4(a1, b[t], acc[1][t]);
        }
    }

    const float scale = 1.0f / (float)B_DIM;
#pragma unroll
    for (int i = 0; i < 2; ++i) {
#pragma unroll
        for (int t = 0; t < 4; ++t) {
#pragma unroll
            for (int r = 0; r < 8; ++r) {
                const int m = mtile + i * 16 + r + half * 8;
                const int n = nbase + t * 16 + l16;
                if (m < NUM2 && n < NUM1) sim[m * SIM_LD + n] = acc[i][t][r] * scale;
            }
        }
    }
}

// ---------------------------------------------------------------------------
// Kernel 2: per row m, select top-200 of 1000 by rank counting, then
// w2[m][h] = mean over selected n of w[fidx1[n]][h]. One block per row.
// ---------------------------------------------------------------------------
__global__ __launch_bounds__(256) void topk_w2_kernel(const float* __restrict__ sim,
                                                      const float* __restrict__ w,
                                                      const int* __restrict__ fidx1,
                                                      float* __restrict__ w2) {
    __shared__ float row[NUM1];
    __shared__ int   sel[NUM1];
    const int m   = blockIdx.x;
    const int tid = threadIdx.x;

    for (int n = tid; n < NUM1; n += 256) row[n] = sim[m * SIM_LD + n];
    __syncthreads();

    for (int n = tid; n < NUM1; n += 256) {
        const float v = row[n];
        int cnt = 0;
        for (int j = 0; j < NUM1; ++j) {
            const float u = row[j];
            cnt += (u > v || (u == v && j < n)) ? 1 : 0;
        }
        sel[n] = (cnt < TOPK) ? 1 : 0;
    }
    __syncthreads();

    const int h   = tid;  // HID == blockDim
    float     acc = 0.0f;
    for (int n = 0; n < NUM1; ++n) {
        if (sel[n]) acc += w[(size_t)fidx1[n] * HID + h];
    }
    w2[(size_t)m * HID + h] = acc * (1.0f / (float)TOPK);
}

// ---------------------------------------------------------------------------
// Kernel 3: wcat[j][h] = j<1000 ? w[fidx1[j]][h] : w2[j-1000][h];
//           catidx[j]  = j<1000 ? fidx1[j] : fidx2[j-1000]
// ---------------------------------------------------------------------------
__global__ __launch_bounds__(256) void build_wcat(const float* __restrict__ w,
                                                  const float* __restrict__ w2,
                                                  const int* __restrict__ fidx1,
                                                  const int* __restrict__ fidx2,
                                                  float* __restrict__ wcat,
                                                  int* __restrict__ catidx) {
    const int j = blockIdx.x;
    const int h = threadIdx.x;
    if (j < NUM1)
        wcat[(size_t)j * HID + h] = w[(size_t)fidx1[j] * HID + h];
    else
        wcat[(size_t)j * HID + h] = w2[(size_t)(j - NUM1) * HID + h];
    if (h == 0) catidx[j] = (j < NUM1) ? fidx1[j] : fidx2[j - NUM1];
}

// ---------------------------------------------------------------------------
// Kernel 4: h0 = x_ @ wcat   (M=16384, K=1200, N=256), gather via catidx (LDS).
// Wave tile = 32(M) x 64(N); block = 4 waves stacked in M (128 rows/block).
// ---------------------------------------------------------------------------
__global__ __launch_bounds__(128) void gemm_xw(const float* __restrict__ x,
                                               const int* __restrict__ catidx,
                                               const float* __restrict__ wcat,
                                               float* __restrict__ h0) {
    __shared__ int scat[FEAT];
    for (int i = threadIdx.x; i < FEAT; i += 128) scat[i] = catidx[i];
    __syncthreads();

    const int lane  = threadIdx.x & 31;
    const int wave  = threadIdx.x >> 5;
    const int l16   = lane & 15;
    const int half  = lane >> 4;
    const int mtile = (blockIdx.y * 4 + wave) * 32;
    const int nbase = blockIdx.x * 64;

    const float* __restrict__ arow0p = x + (size_t)(mtile + l16) * FEAT;
    const float* __restrict__ arow1p = arow0p + (size_t)16 * FEAT;

    v8f acc[2][4] = {};
#pragma unroll 2
    for (int k0 = 0; k0 < FEAT; k0 += 4) {
        const int ka = k0 + half * 2;
        const int c0 = scat[ka], c1 = scat[ka + 1];
        v2f a0, a1;
        a0.x = arow0p[c0]; a0.y = arow0p[c1];
        a1.x = arow1p[c0]; a1.y = arow1p[c1];
        const float* __restrict__ b0p = wcat + (size_t)ka * HID + nbase + l16;
#pragma unroll
        for (int t = 0; t < 4; ++t) {
            v2f b;
            b.x = b0p[t * 16];
            b.y = b0p[t * 16 + HID];
            acc[0][t] = wmma_f32_k4(a0, b, acc[0][t]);
            acc[1][t] = wmma_f32_k4(a1, b, acc[1][t]);
        }
    }
#pragma unroll
    for (int i = 0; i < 2; ++i) {
#pragma unroll
        for (int t = 0; t < 4; ++t) {
#pragma unroll
            for (int r = 0; r < 8; ++r) {
                const size_t mrow = (size_t)(mtile + i * 16 + r + half * 8);
                h0[mrow * HID + nbase + t * 16 + l16] = acc[i][t][r];
            }
        }
    }
}

// ---------------------------------------------------------------------------
// Kernel 5: hout = relu(hin @ W + bias)   (M=16384, K=256, N=256)
// Wave tile = 32(M) x 64(N); A fragments loaded as b64 pairs.
// ---------------------------------------------------------------------------
__global__ __launch_bounds__(128) void gemm_hidden(const float* __restrict__ hin,
                                                   const float* __restrict__ W,
                                                   const float* __restrict__ bias,
                                                   float* __restrict__ hout) {
    const int lane  = threadIdx.x & 31;
    const int wave  = threadIdx.x >> 5;
    const int l16   = lane & 15;
    const int half  = lane >> 4;
    const int mtile = (blockIdx.y * 4 + wave) * 32;
    const int nbase = blockIdx.x * 64;

    const float* __restrict__ arow0p = hin + (size_t)(mtile + l16) * HID;
    const float* __restrict__ arow1p = arow0p + (size_t)16 * HID;

    v8f acc[2][4] = {};
#pragma unroll 2
    for (int k0 = 0; k0 < HID; k0 += 4) {
        const int ka = k0 + half * 2;
        const v2f a0 = *(const v2f*)(arow0p + ka);
        const v2f a1 = *(const v2f*)(arow1p + ka);
        const float* __restrict__ b0p = W + (size_t)ka * HID + nbase + l16;
#pragma unroll
        for (int t = 0; t < 4; ++t) {
            v2f b;
            b.x = b0p[t * 16];
            b.y = b0p[t * 16 + HID];
            acc[0][t] = wmma_f32_k4(a0, b, acc[0][t]);
            acc[1][t] = wmma_f32_k4(a1, b, acc[1][t]);
        }
    }
#pragma unroll
    for (int i = 0; i < 2; ++i) {
#pragma unroll
        for (int t = 0; t < 4; ++t) {
            const float bv = bias[nbase + t * 16 + l16];
#pragma unroll
            for (int r = 0; r < 8; ++r) {
                const size_t mrow = (size_t)(mtile + i * 16 + r + half * 8);
                hout[mrow * HID + nbase + t * 16 + l16] = fmaxf(acc[i][t][r] + bv, 0.0f);
            }
        }
    }
}

// ---------------------------------------------------------------------------
// Kernel 6: out = hin @ l3_w + l3_b   (M=16384, K=256, N=10 padded to 16).
// Wave tile = 32(M) x 16(N); B column index clamped (cols 10..15 never stored).
// ---------------------------------------------------------------------------
__global__ __launch_bounds__(128) void gemm_out(const float* __restrict__ hin,
                                                const float* __restrict__ W,
                                                const float* __restrict__ bias,
                                                float* __restrict__ out) {
    const int lane  = threadIdx.x & 31;
    const int wave  = threadIdx.x >> 5;
    const int l16   = lane & 15;
    const int half  = lane >> 4;
    const int mtile = (blockIdx.x * 4 + wave) * 32;

    const float* __restrict__ arow0p = hin + (size_t)(mtile + l16) * HID;
    const float* __restrict__ arow1p = arow0p + (size_t)16 * HID;
    const int colB = imin(l16, CLS - 1);  // clamped -> unconditional loads

    v8f acc[2] = {};
#pragma unroll 2
    for (int k0 = 0; k0 < HID; k0 += 4) {
        const int ka = k0 + half * 2;
        const v2f a0 = *(const v2f*)(arow0p + ka);
        const v2f a1 = *(const v2f*)(arow1p + ka);
        v2f b;
        b.x = W[(size_t)(ka    ) * CLS + colB];
        b.y = W[(size_t)(ka + 1) * CLS + colB];
        acc[0] = wmma_f32_k4(a0, b, acc[0]);
        acc[1] = wmma_f32_k4(a1, b, acc[1]);
    }
    const float bv = bias[colB];
    if (l16 < CLS) {
#pragma unroll
        for (int i = 0; i < 2; ++i) {
#pragma unroll
            for (int r = 0; r < 8; ++r) {
                const size_t mrow = (size_t)(mtile + i * 16 + r + half * 8);
                out[mrow * CLS + l16] = acc[i][r] + bv;
            }
        }
    }
}

// ---------------------------------------------------------------------------
extern "C" void kernel_launch(void* const* d_in, const int* in_sizes, int n_in,
                              void* d_out, int out_size, void* d_ws, size_t ws_size,
                              hipStream_t stream) {
    const float* x     = (const float*)d_in[0];
    const float* w     = (const float*)d_in[1];
    const float* l1_w  = (const float*)d_in[2];
    const float* l1_b  = (const float*)d_in[3];
    const float* l2_w  = (const float*)d_in[4];
    const float* l2_b  = (const float*)d_in[5];
    const float* l3_w  = (const float*)d_in[6];
    const float* l3_b  = (const float*)d_in[7];
    const int*   fidx1 = (const int*)d_in[8];
    const int*   fidx2 = (const int*)d_in[9];
    float*       out   = (float*)d_out;

    // Workspace layout (floats): sim | w2 | wcat | catidx | h0 | h1  (~36 MB)
    float* ws     = (float*)d_ws;
    float* sim    = ws;                                   // 200*1024
    float* w2     = sim + NUM2 * SIM_LD;                  // 200*256
    float* wcat   = w2 + NUM2 * HID;                      // 1200*256
    int*   catidx = (int*)(wcat + FEAT * HID);            // 1200 (+pad)
    float* h0     = (float*)(catidx + 1280);              // 16384*256
    float* h1     = h0 + (size_t)B_DIM * HID;             // 16384*256

    // sim: M padded to 7x32=224 rows, N: 16 groups of 64 (covers 1000, clamped)
    sim_kernel<<<dim3(4, 7), 128, 0, stream>>>(x, fidx1, fidx2, sim);
    topk_w2_kernel<<<NUM2, 256, 0, stream>>>(sim, w, fidx1, w2);
    build_wcat<<<FEAT, 256, 0, stream>>>(w, w2, fidx1, fidx2, wcat, catidx);
    gemm_xw<<<dim3(HID / 64, B_DIM / 128), 128, 0, stream>>>(x, catidx, wcat, h0);
    gemm_hidden<<<dim3(HID / 64, B_DIM / 128), 128, 0, stream>>>(h0, l1_w, l1_b, h1);
    gemm_hidden<<<dim3(HID / 64, B_DIM / 128), 128, 0, stream>>>(h1, l2_w, l2_b, h0);
    gemm_out<<<B_DIM / 128, 128, 0, stream>>>(h0, l3_w, l3_b, out);
}